// HistogramLoss_77807627534942
// MI455X (gfx1250) — compile-verified
//
#include <hip/hip_runtime.h>
#include <hip/hip_bf16.h>

#define H    2048
#define HH   (H * H)
#define BINS 256

typedef __attribute__((ext_vector_type(2))) float v2f;
typedef __attribute__((ext_vector_type(4))) float v4f;
typedef __attribute__((ext_vector_type(8))) float v8f;

__device__ __forceinline__ float denorm255(float x) {
    float t = (x + 1.0f) * 0.5f;            // (x+1)/2  (exact: /2 == *0.5)
    t = fminf(fmaxf(t, 0.0f), 1.0f);        // clip to [0,1]
    return t * 255.0f;
}

// ---------------------------------------------------------------- zero ws
__global__ void zero_ws_kernel(unsigned int* __restrict__ w, int nwords) {
    for (int i = blockIdx.x * blockDim.x + threadIdx.x; i < nwords;
         i += gridDim.x * blockDim.x)
        w[i] = 0u;
}

// ---------------------------------------------------------------- pass 1: 6 histograms
// hist[0..2] = floor(tgt * m_src) at m_src!=0 ; hist[3..5] = floor(denorm255(mkp)*m_tar) at m_tar!=0
__global__ __launch_bounds__(256) void hist_kernel(
    const float* __restrict__ tgt, const float* __restrict__ mkp,
    const float* __restrict__ msrc, const float* __restrict__ mtar,
    unsigned int* __restrict__ ghist)
{
    __shared__ unsigned int lh[6 * BINS];
    for (int t = threadIdx.x; t < 6 * BINS; t += blockDim.x) lh[t] = 0u;
    __syncthreads();

    const int HH4 = HH / 4;
    const v4f* ms4 = (const v4f*)msrc;
    const v4f* mt4 = (const v4f*)mtar;

    for (int i = blockIdx.x * blockDim.x + threadIdx.x; i < HH4;
         i += gridDim.x * blockDim.x) {
        v4f ms = ms4[i], mt = mt4[i];   // masks: keep regular-temporal (re-read in pass 3)
        bool anyS = (ms[0] != 0.f) | (ms[1] != 0.f) | (ms[2] != 0.f) | (ms[3] != 0.f);
        bool anyT = (mt[0] != 0.f) | (mt[1] != 0.f) | (mt[2] != 0.f) | (mt[3] != 0.f);

        if (anyS) {
            #pragma unroll
            for (int c = 0; c < 3; ++c) {
                // tgt is re-read by the loss pass: regular-temporal load (L2-resident)
                v4f t4 = ((const v4f*)(tgt + (size_t)c * HH))[i];
                #pragma unroll
                for (int e = 0; e < 4; ++e) {
                    if (ms[e] != 0.f) {
                        float v = t4[e] * ms[e];
                        int b = (int)floorf(v);
                        b = min(max(b, 0), BINS - 1);
                        atomicAdd(&lh[c * BINS + b], 1u);   // ds_add_u32
                    }
                }
            }
        }
        if (anyT) {
            #pragma unroll
            for (int c = 0; c < 3; ++c) {
                // mkp is single-use: non-temporal so it doesn't evict tgt/masks from L2
                v4f k4 = __builtin_nontemporal_load(
                             (const v4f*)(mkp + (size_t)c * HH) + i);
                #pragma unroll
                for (int e = 0; e < 4; ++e) {
                    if (mt[e] != 0.f) {
                        float v = denorm255(k4[e]) * mt[e];
                        int b = (int)floorf(v);
                        b = min(max(b, 0), BINS - 1);
                        atomicAdd(&lh[(3 + c) * BINS + b], 1u);
                    }
                }
            }
        }
    }
    __syncthreads();
    for (int t = threadIdx.x; t < 6 * BINS; t += blockDim.x) {
        unsigned int v = lh[t];
        if (v) atomicAdd(&ghist[t], v);
    }
}

// ---------------------------------------------------------------- pass 2: CDFs + transfer tables
__global__ __launch_bounds__(256) void table_kernel(
    const unsigned int* __restrict__ ghist, float* __restrict__ tables)
{
    __shared__ float cdf[6 * BINS];
    const int tid = threadIdx.x;

    // sequential left-to-right cumsum of pdf = hist/total (matches jnp f32 semantics;
    // counts < 2^24 so the f32 total is exact)
    if (tid < 6) {
        const unsigned int* h = ghist + tid * BINS;
        unsigned int total = 0;
        for (int i = 0; i < BINS; ++i) total += h[i];
        float ftot = (float)total;
        float run = 0.0f;
        for (int i = 0; i < BINS; ++i) {
            run += (float)h[i] / ftot;
            cdf[tid * BINS + i] = run;
        }
    }
    __syncthreads();

    for (int c = 0; c < 3; ++c) {
        const float* cd = &cdf[c * BINS];
        const float* cr = &cdf[(3 + c) * BINS];
        float d = cd[tid];
        int tbl = tid;
        if (tid == BINS - 1) {
            tbl = BINS - 1;                     // reference forces table[255] = 255
        } else {
            for (int j = 0; j < BINS - 1; ++j) {
                if (d >= cr[j] && d <= cr[j + 1]) { tbl = j + 1; break; }
            }
        }
        tables[c * BINS + tid] = (float)tbl;
    }
}

// ---------------------------------------------------------------- pass 3: loss sum
__global__ __launch_bounds__(256) void loss_kernel(
    const float* __restrict__ inp, const float* __restrict__ tgt,
    const float* __restrict__ msrc, const float* __restrict__ tables,
    double* __restrict__ acc)
{
    __shared__ float tbl[3 * BINS];
    __shared__ float sdata[256];
    for (int t = threadIdx.x; t < 3 * BINS; t += blockDim.x) tbl[t] = tables[t];
    __syncthreads();

    float s = 0.0f;
    const int HH4 = HH / 4;
    const v4f* ms4 = (const v4f*)msrc;

    for (int i = blockIdx.x * blockDim.x + threadIdx.x; i < HH4;
         i += gridDim.x * blockDim.x) {
        v4f ms = ms4[i];                    // mask: should hit L2 from pass 1
        if ((ms[0] != 0.f) | (ms[1] != 0.f) | (ms[2] != 0.f) | (ms[3] != 0.f)) {
            #pragma unroll
            for (int c = 0; c < 3; ++c) {
                // inp is single-use: non-temporal; tgt should hit L2 from pass 1
                v4f i4 = __builtin_nontemporal_load(
                             (const v4f*)(inp + (size_t)c * HH) + i);
                v4f t4 = ((const v4f*)(tgt + (size_t)c * HH))[i];
                #pragma unroll
                for (int e = 0; e < 4; ++e) {
                    if (ms[e] != 0.f) {
                        float im  = denorm255(i4[e]) * ms[e];
                        float dv  = t4[e] * ms[e];
                        int  idx  = (int)dv;                    // astype(int32): trunc
                        idx = min(max(idx, 0), BINS - 1);
                        s += fabsf(im - tbl[c * BINS + idx]);
                    }
                }
            }
        }
    }

    sdata[threadIdx.x] = s;
    __syncthreads();

    // Wave 0 (threads 0..31 == one full wave32, EXEC all ones) reduces 256 partials.
    if (threadIdx.x < 32) {
        const int lane = threadIdx.x;
        float r;
#if defined(__gfx1250__) && __has_builtin(__builtin_amdgcn_wmma_f32_16x16x4_f32)
        // D += Ones(16x4) x B(4x16), 4 slices covering all 256 values.
        // With A == all-ones, every fed value is summed into some column of D
        // independent of the B lane->(k,j) mapping; all 16 rows of D are equal
        // (= column sums), so VGPR0 across 32 lanes holds each column sum twice:
        // sum(lanes of d[0]) == 2 * total.
        v2f a; a[0] = 1.0f; a[1] = 1.0f;
        v8f d = {};
        #pragma unroll
        for (int sl = 0; sl < 4; ++sl) {
            v2f b;
            b[0] = sdata[lane + 64 * sl];
            b[1] = sdata[lane + 64 * sl + 32];
            d = __builtin_amdgcn_wmma_f32_16x16x4_f32(
                    false, a, false, b, (short)0, d, false, false);
        }
        r = d[0];
        #pragma unroll
        for (int off = 16; off > 0; off >>= 1) r += __shfl_xor(r, off, 32);
        r *= 0.5f;
#else
        r = 0.0f;
        for (int k = lane; k < 256; k += 32) r += sdata[k];
        #pragma unroll
        for (int off = 16; off > 0; off >>= 1) r += __shfl_xor(r, off, 32);
#endif
        if (lane == 0) {
#if defined(__gfx1250__)
            unsafeAtomicAdd(acc, (double)r);    // native global_atomic_add_f64
#else
            atomicAdd(acc, (double)r);
#endif
        }
    }
}

// ---------------------------------------------------------------- finalize
__global__ void finalize_kernel(const double* __restrict__ acc, float* __restrict__ out) {
    out[0] = (float)(acc[0] / (double)(3.0 * (double)HH));
}

// ---------------------------------------------------------------- launch
extern "C" void kernel_launch(void* const* d_in, const int* in_sizes, int n_in,
                              void* d_out, int out_size, void* d_ws, size_t ws_size,
                              hipStream_t stream) {
    (void)in_sizes; (void)n_in; (void)out_size; (void)ws_size;
    const float* inp  = (const float*)d_in[0];   // input_data  (3*H*H)
    const float* tgt  = (const float*)d_in[1];   // target_data (3*H*H)
    const float* mkp  = (const float*)d_in[2];   // makeup_data (3*H*H)
    const float* msrc = (const float*)d_in[3];   // mask_src (H*H)
    const float* mtar = (const float*)d_in[4];   // mask_tar (H*H)
    // d_in[5..8] = x_A,y_A,x_B,y_B — redundant (they enumerate the mask nonzeros)

    unsigned int* ghist  = (unsigned int*)d_ws;                                  // 6*256 u32
    float*        tables = (float*)((char*)d_ws + 6 * BINS * sizeof(unsigned));  // 3*256 f32
    double*       acc    = (double*)((char*)d_ws + 9 * BINS * sizeof(unsigned)); // offset 9216, 8B aligned

    zero_ws_kernel<<<1, 256, 0, stream>>>((unsigned int*)d_ws, 6 * BINS + 3 * BINS + 2);
    hist_kernel  <<<1024, 256, 0, stream>>>(tgt, mkp, msrc, mtar, ghist);
    table_kernel <<<1, 256, 0, stream>>>(ghist, tables);
    loss_kernel  <<<1024, 256, 0, stream>>>(inp, tgt, msrc, tables, acc);
    finalize_kernel<<<1, 1, 0, stream>>>(acc, (float*)d_out);
}